// GNNModel_67817533604361
// MI455X (gfx1250) — compile-verified
//
#include <hip/hip_runtime.h>

#define N_NODES 50000
#define N_EDGES 800000
#define FEAT 64
#define HID 128
#define N_REL 3
#define LN_EPS 1e-5f
#define ASTRIDE 132  // padded LDS row stride (floats): conflict-free & 16B-aligned rows

typedef float v2f __attribute__((ext_vector_type(2)));
typedef float v8f __attribute__((ext_vector_type(8)));

__global__ void zero_kernel(float* __restrict__ p, long long n) {
  long long i = (long long)blockIdx.x * blockDim.x + threadIdx.x;
  long long stride = (long long)gridDim.x * blockDim.x;
  for (; i < n; i += stride) p[i] = 0.0f;
}

__global__ void deg_kernel(const int* __restrict__ dst, float* __restrict__ deg, int E) {
  int e = blockIdx.x * blockDim.x + threadIdx.x;
  if (e < E) atomicAdd(&deg[dst[e]], 1.0f);
}

// One block (128 threads) per node: type-specific 2-layer MLP encoder.
__global__ void encoder_kernel(const int* __restrict__ z, const int* __restrict__ ntype,
                               const float* __restrict__ z_embed,
                               const float* __restrict__ w1, const float* __restrict__ b1,
                               const float* __restrict__ w2, const float* __restrict__ b2,
                               float* __restrict__ x) {
  __shared__ float raw[FEAT];
  __shared__ float hid[HID];
  int n = blockIdx.x;
  int t = threadIdx.x;
  int ty = ntype[n];
  if (t < FEAT) raw[t] = z_embed[z[n] * FEAT + t];
  __syncthreads();
  const float* W1 = w1 + (size_t)ty * FEAT * HID;
  float acc = b1[ty * HID + t];
#pragma unroll 8
  for (int d = 0; d < FEAT; ++d) acc = fmaf(raw[d], W1[d * HID + t], acc);
  hid[t] = fmaxf(acc, 0.0f);
  __syncthreads();
  const float* W2 = w2 + (size_t)ty * HID * HID;
  float acc2 = b2[ty * HID + t];
#pragma unroll 8
  for (int d = 0; d < HID; ++d) acc2 = fmaf(hid[d], W2[d * HID + t], acc2);
  x[(size_t)n * HID + t] = acc2;
}

// One wave per edge: agg[etype][dst][c] += x[src][c]
__global__ void scatter_kernel(const int* __restrict__ src, const int* __restrict__ dst,
                               const int* __restrict__ etype,
                               const float* __restrict__ x, float* __restrict__ agg, int E) {
  int gid = blockIdx.x * blockDim.x + threadIdx.x;
  int e = gid >> 5;
  int lane = gid & 31;
  if (e >= E) return;
  int s = src[e], d = dst[e], r = etype[e];
  const float* xs = x + (size_t)s * HID;
  float* ag = agg + ((size_t)r * N_NODES + d) * HID;
#pragma unroll
  for (int i = 0; i < HID / 32; ++i) {
    int c = lane + i * 32;
    atomicAdd(&ag[c], xs[c]);
  }
}

// 256 threads = 8 waves per 16-node tile. Each wave owns a 16x16 output tile.
// 4 merged GEMMs over K=128:  g=0: xin@lin_w   g=1..3: agg[r]@rel_w[r]
// A tiles staged in LDS (shared by all 8 waves), B double-buffered from L2.
__global__ void __launch_bounds__(256) combine_kernel(
    const float* __restrict__ xin, const float* __restrict__ agg,
    const float* __restrict__ deg,
    const float* __restrict__ lin_w, const float* __restrict__ lin_b,
    const float* __restrict__ rel_w,
    const float* __restrict__ ln_g, const float* __restrict__ ln_b,
    float* __restrict__ xout) {
  __shared__ float A4[4 * 16 * ASTRIDE];  // 4 staged A tiles; reused as LN tile
  __shared__ float sdeg[16];
  __shared__ float smu[16], srs[16];

  int m0 = blockIdx.x * 16;
  int tid = threadIdx.x;
  int lane = tid & 31;
  int wave = tid >> 5;
  int n0 = wave * 16;
  int half = lane >> 4;      // 0 or 1 (lane group)
  int mrow = lane & 15;      // A-fragment row / B-fragment column
  int khalf = half * 2;      // K offset within a 4-wide step

  if (tid < 16) sdeg[tid] = 1.0f / fmaxf(deg[m0 + tid], 1.0f);

  // ---- cooperative A staging: 4 tiles x 16 rows x 32 float4 (coalesced) ----
  for (int i = tid; i < 4 * 16 * 32; i += 256) {
    int mat = i >> 9;          // /512: which matrix
    int rem = i & 511;
    int m = rem >> 5;          // row 0..15
    int q = rem & 31;          // float4 within row
    const float* srcp = (mat == 0)
        ? xin + (size_t)(m0 + m) * HID
        : agg + ((size_t)(mat - 1) * N_NODES + m0 + m) * HID;
    float4 v = *(const float4*)(srcp + q * 4);
    *(float4*)&A4[((mat * 16) + m) * ASTRIDE + q * 4] = v;
  }
  __syncthreads();

  // ---- merged, pipelined GEMMs ----
  const float* Wg[4];
  Wg[0] = lin_w;
  Wg[1] = rel_w;
  Wg[2] = rel_w + HID * HID;
  Wg[3] = rel_w + 2 * HID * HID;

  int coff = n0 + mrow;  // this lane's B column / output channel

  v2f bf[4];
#pragma unroll
  for (int g = 0; g < 4; ++g) {
    bf[g].x = Wg[g][khalf * HID + coff];
    bf[g].y = Wg[g][(khalf + 1) * HID + coff];
  }

  v8f acc0 = {}, acc1 = {}, acc2 = {}, acc3 = {};
#pragma unroll
  for (int kk = 0; kk < 32; ++kk) {
    v2f bn[4];
    if (kk < 31) {  // prefetch next-step B fragments (independent regs)
      int kN = (kk + 1) * 4 + khalf;
#pragma unroll
      for (int g = 0; g < 4; ++g) {
        bn[g].x = Wg[g][kN * HID + coff];
        bn[g].y = Wg[g][(kN + 1) * HID + coff];
      }
    }
    int kA = kk * 4 + khalf;
    v2f a0 = *(const v2f*)&A4[(0 * 16 + mrow) * ASTRIDE + kA];
    v2f a1 = *(const v2f*)&A4[(1 * 16 + mrow) * ASTRIDE + kA];
    v2f a2 = *(const v2f*)&A4[(2 * 16 + mrow) * ASTRIDE + kA];
    v2f a3 = *(const v2f*)&A4[(3 * 16 + mrow) * ASTRIDE + kA];
    acc0 = __builtin_amdgcn_wmma_f32_16x16x4_f32(false, a0, false, bf[0], (short)0, acc0, false, false);
    acc1 = __builtin_amdgcn_wmma_f32_16x16x4_f32(false, a1, false, bf[1], (short)0, acc1, false, false);
    acc2 = __builtin_amdgcn_wmma_f32_16x16x4_f32(false, a2, false, bf[2], (short)0, acc2, false, false);
    acc3 = __builtin_amdgcn_wmma_f32_16x16x4_f32(false, a3, false, bf[3], (short)0, acc3, false, false);
#pragma unroll
    for (int g = 0; g < 4; ++g) bf[g] = bn[g];
  }

  __syncthreads();  // all A reads done; reuse A4[0..] as the LN staging tile
  float* tile = A4;
  int mb = half * 8;  // C/D layout: VGPR v -> row v + 8*(lane>>4)
#pragma unroll
  for (int v = 0; v < 8; ++v) {
    int m = mb + v;
    tile[m * ASTRIDE + coff] =
        acc0[v] + lin_b[coff] + (acc1[v] + acc2[v] + acc3[v]) * sdeg[m];
  }
  __syncthreads();

  if (tid < 16) {
    float s = 0.0f, s2 = 0.0f;
    for (int i = 0; i < HID; ++i) {
      float v = tile[tid * ASTRIDE + i];
      s += v; s2 += v * v;
    }
    float mu = s * (1.0f / HID);
    float var = s2 * (1.0f / HID) - mu * mu;
    smu[tid] = mu;
    srs[tid] = rsqrtf(var + LN_EPS);
  }
  __syncthreads();

  for (int i = tid; i < 16 * HID; i += 256) {
    int m = i >> 7;
    int cc = i & 127;
    xout[(size_t)(m0 + m) * HID + cc] =
        (tile[m * ASTRIDE + cc] - smu[m]) * srs[m] * ln_g[cc] + ln_b[cc];
  }
}

__global__ void pool_partial(const float* __restrict__ x, float* __restrict__ pooled) {
  int c = threadIdx.x;
  float local = 0.0f;
  for (int n = blockIdx.x; n < N_NODES; n += gridDim.x)
    local += x[(size_t)n * HID + c];
  atomicAdd(&pooled[c], local);
}

__global__ void pool_final(const float* __restrict__ pooled, const float* __restrict__ reg_w,
                           const float* __restrict__ reg_b, float* __restrict__ out) {
  __shared__ float red[HID];
  int t = threadIdx.x;
  red[t] = pooled[t] * (1.0f / N_NODES) * reg_w[t];
  __syncthreads();
  for (int s = HID / 2; s > 0; s >>= 1) {
    if (t < s) red[t] += red[t + s];
    __syncthreads();
  }
  if (t == 0) out[0] = red[0] + reg_b[0];
}

extern "C" void kernel_launch(void* const* d_in, const int* in_sizes, int n_in,
                              void* d_out, int out_size, void* d_ws, size_t ws_size,
                              hipStream_t stream) {
  const int* z          = (const int*)d_in[0];
  const int* ntype      = (const int*)d_in[1];
  const int* edge_index = (const int*)d_in[2];
  const int* etype      = (const int*)d_in[3];
  const float* z_embed  = (const float*)d_in[4];
  const float* enc_w1   = (const float*)d_in[5];
  const float* enc_b1   = (const float*)d_in[6];
  const float* enc_w2   = (const float*)d_in[7];
  const float* enc_b2   = (const float*)d_in[8];
  const float* lin_w    = (const float*)d_in[9];
  const float* lin_b    = (const float*)d_in[10];
  const float* rel_w    = (const float*)d_in[11];
  const float* ln_g     = (const float*)d_in[12];
  const float* ln_b     = (const float*)d_in[13];
  const float* reg_w    = (const float*)d_in[14];
  const float* reg_b    = (const float*)d_in[15];

  float* ws = (float*)d_ws;
  float* x0     = ws;                               // N*H
  float* x1     = x0 + (size_t)N_NODES * HID;       // N*H
  float* deg    = x1 + (size_t)N_NODES * HID;       // N
  float* pooled = deg + N_NODES;                    // H
  float* agg    = pooled + HID;                     // 3*N*H

  const int* src = edge_index;
  const int* dst = edge_index + N_EDGES;

  // zero deg + pooled (contiguous region)
  zero_kernel<<<256, 256, 0, stream>>>(deg, (long long)N_NODES + HID);
  deg_kernel<<<(N_EDGES + 255) / 256, 256, 0, stream>>>(dst, deg, N_EDGES);
  encoder_kernel<<<N_NODES, HID, 0, stream>>>(z, ntype, z_embed, enc_w1, enc_b1,
                                              enc_w2, enc_b2, x0);

  for (int l = 0; l < 2; ++l) {
    const float* xin = (l == 0) ? x0 : x1;
    float* xout      = (l == 0) ? x1 : x0;
    zero_kernel<<<2048, 256, 0, stream>>>(agg, (long long)N_REL * N_NODES * HID);
    scatter_kernel<<<(N_EDGES * 32 + 255) / 256, 256, 0, stream>>>(src, dst, etype,
                                                                   xin, agg, N_EDGES);
    combine_kernel<<<N_NODES / 16, 256, 0, stream>>>(
        xin, agg, deg,
        lin_w + (size_t)l * HID * HID, lin_b + l * HID,
        rel_w + (size_t)l * N_REL * HID * HID,
        ln_g + l * HID, ln_b + l * HID, xout);
  }

  pool_partial<<<256, HID, 0, stream>>>(x0, pooled);
  pool_final<<<1, HID, 0, stream>>>(pooled, reg_w, reg_b, (float*)d_out);
}